// SecondaryStatistic_75711683494098
// MI455X (gfx1250) — compile-verified
//
#include <hip/hip_runtime.h>
#include <stdint.h>

typedef float v2f  __attribute__((ext_vector_type(2)));
typedef float v8f  __attribute__((ext_vector_type(8)));
typedef unsigned int u32x4 __attribute__((ext_vector_type(4)));
typedef int   i32x4 __attribute__((ext_vector_type(4)));
typedef int   i32x8 __attribute__((ext_vector_type(8)));

#define N_BATCH 64
#define F_DIM   1024
#define HW      256
#define BM      128          // output tile is BM x BM
#define FSTR    136          // padded LDS panel row stride in floats (128 + 8)
#define SSTR    132          // padded LDS stage row stride (mirror transpose)
#define EPS_I   1e-5f

// ---- Tensor Data Mover: 2D tile (tile_dim0=128 f-contig, tile_dim1=256 rows,
// ---- row stride F_DIM) -> LDS with pad 8 DWORDs every 128 DWORDs (=> FSTR).
__device__ __forceinline__ void tdm_load_panel(uint64_t gaddr, uint32_t lds_off) {
  u32x4 g0;
  g0[0] = 1u;                                              // count=1, user descriptor
  g0[1] = lds_off;                                         // lds_addr (bytes)
  g0[2] = (uint32_t)(gaddr & 0xFFFFFFFFu);                 // global_addr[31:0]
  g0[3] = (uint32_t)((gaddr >> 32) & 0x01FFFFFFu)          // global_addr[56:32]
        | 0x80000000u;                                     // type = 2 ("image")
  i32x8 g1;
  g1[0] = (int)((2u << 16)       // data_size = 4 bytes
              | (1u << 20)       // pad_enable
              | (6u << 22)       // pad_interval: 2^(6+1)=128 DWORDs
              | (7u << 25));     // pad_amount: 8 DWORDs
  g1[1] = (int)(((uint32_t)F_DIM & 0xFFFFu) << 16);        // tensor_dim0 lo16
  g1[2] = (int)(((uint32_t)HW & 0xFFFFu) << 16);           // dim0 hi | tensor_dim1 lo16
  g1[3] = (int)((uint32_t)BM << 16);                       // dim1 hi | tile_dim0 = 128
  g1[4] = (int)HW;                                         // tile_dim1 = 256, tile_dim2 = 0
  g1[5] = (int)F_DIM;                                      // tensor_dim0_stride lo32
  g1[6] = 0;
  g1[7] = 0;
  i32x4 gz4 = {0, 0, 0, 0};
  i32x8 gz8 = {0, 0, 0, 0, 0, 0, 0, 0};
  __builtin_amdgcn_tensor_load_to_lds(g0, g1, gz4, gz4, gz8, 0);
}

__global__ __launch_bounds__(256)
void SecondaryStatistic_cov_wmma(const float* __restrict__ x, float* __restrict__ out) {
  const int tj = blockIdx.x;     // column tile
  const int ti = blockIdx.y;     // row tile
  const int b  = blockIdx.z;     // batch
  if (tj < ti) return;           // symmetric: only upper-triangular tile pairs

  __shared__ float lds[2 * HW * FSTR];   // panel A then panel B, [s][f] padded
  __shared__ float muA[BM];
  __shared__ float muB[BM];
  float* ldsA = lds;
  float* ldsB = lds + HW * FSTR;

  const int tid  = threadIdx.x;
  const int lane = tid & 31;
  const int wave = tid >> 5;
  const int hi   = lane >> 4;    // half-wave selector for WMMA frag K offset
  const int l16  = lane & 15;

  const int i0 = ti * BM;
  const int j0 = tj * BM;

  // ---------------- async tensor loads (wave 0 issues, TENSORcnt sync) -------
  if (wave == 0) {
    const uint64_t gA = (uint64_t)(uintptr_t)(x + (size_t)b * HW * F_DIM + i0);
    const uint64_t gB = (uint64_t)(uintptr_t)(x + (size_t)b * HW * F_DIM + j0);
    tdm_load_panel(gA, (uint32_t)(uintptr_t)ldsA);
    tdm_load_panel(gB, (uint32_t)(uintptr_t)ldsB);
    __builtin_amdgcn_s_wait_tensorcnt(0);
  }
  __syncthreads();

  // ---------------- per-feature spatial means from LDS panels ----------------
  {
    const float* p = (tid < BM) ? ldsA : ldsB;
    const int f = tid & (BM - 1);
    float s = 0.0f;
#pragma unroll 8
    for (int sp = 0; sp < HW; ++sp) s += p[sp * FSTR + f];
    const float mu = s * (1.0f / (float)HW);
    if (tid < BM) muA[f] = mu; else muB[f] = mu;
  }
  __syncthreads();

  // ---------------- WMMA main loop: G = Xi * Xj^T over K=256 -----------------
  const int wm = wave & 3;   // 4 wave-rows  -> 32 rows each
  const int wn = wave >> 2;  // 2 wave-cols  -> 64 cols each

  v8f acc[2][4];
#pragma unroll
  for (int mt = 0; mt < 2; ++mt)
#pragma unroll
    for (int nt = 0; nt < 4; ++nt) {
      v8f z = {0.f, 0.f, 0.f, 0.f, 0.f, 0.f, 0.f, 0.f};
      acc[mt][nt] = z;
    }

  const float* aBase = ldsA + (wm * 32 + l16);
  const float* bBase = ldsB + (wn * 64 + l16);

#pragma unroll 2
  for (int k = 0; k < HW; k += 4) {
    const int s0 = (k + 2 * hi) * FSTR;
    v2f afr[2], bfr[4];
#pragma unroll
    for (int mt = 0; mt < 2; ++mt) {
      afr[mt][0] = aBase[s0 + mt * 16];
      afr[mt][1] = aBase[s0 + FSTR + mt * 16];
    }
#pragma unroll
    for (int nt = 0; nt < 4; ++nt) {
      bfr[nt][0] = bBase[s0 + nt * 16];
      bfr[nt][1] = bBase[s0 + FSTR + nt * 16];
    }
#pragma unroll
    for (int mt = 0; mt < 2; ++mt)
#pragma unroll
      for (int nt = 0; nt < 4; ++nt)
        acc[mt][nt] = __builtin_amdgcn_wmma_f32_16x16x4_f32(
            false, afr[mt], false, bfr[nt], (short)0, acc[mt][nt], false, false);
  }

  // ---------------- epilogue: mean correction + eps*I, coalesced store -------
  const float inv = 1.0f / (float)(HW - 1);
#pragma unroll
  for (int mt = 0; mt < 2; ++mt) {
#pragma unroll
    for (int nt = 0; nt < 4; ++nt) {
      const int n128 = wn * 64 + nt * 16 + l16;
      const float mub = muB[n128];
#pragma unroll
      for (int v = 0; v < 8; ++v) {
        const int m128 = wm * 32 + mt * 16 + v + 8 * hi;
        float c = (acc[mt][nt][v] - (float)HW * muA[m128] * mub) * inv;
        if (i0 + m128 == j0 + n128) c += EPS_I;
        acc[mt][nt][v] = c;
        out[((size_t)b * F_DIM + (i0 + m128)) * F_DIM + (j0 + n128)] = c;
      }
    }
  }

  // ---------------- mirror tile via LDS transpose (coalesced) ----------------
  if (ti != tj) {
    __syncthreads();                 // panels no longer needed; reuse as stage
    float* stage = lds;              // BM x SSTR
#pragma unroll
    for (int mt = 0; mt < 2; ++mt)
#pragma unroll
      for (int nt = 0; nt < 4; ++nt) {
        const int n128 = wn * 64 + nt * 16 + l16;
#pragma unroll
        for (int v = 0; v < 8; ++v) {
          const int m128 = wm * 32 + mt * 16 + v + 8 * hi;
          stage[n128 * SSTR + m128] = acc[mt][nt][v];
        }
      }
    __syncthreads();
    // out[b][j0+r][i0+c] = stage[r][c], 16B vectorized, fully coalesced
    for (int idx = tid; idx < BM * (BM / 4); idx += 256) {
      const int r  = idx >> 5;       // 32 float4 per row
      const int c4 = (idx & 31) * 4;
      const float4 vdat = *(const float4*)(stage + r * SSTR + c4);
      *(float4*)(out + ((size_t)b * F_DIM + (j0 + r)) * F_DIM + (i0 + c4)) = vdat;
    }
  }
}

extern "C" void kernel_launch(void* const* d_in, const int* in_sizes, int n_in,
                              void* d_out, int out_size, void* d_ws, size_t ws_size,
                              hipStream_t stream) {
  const float* x = (const float*)d_in[0];
  float* out = (float*)d_out;
  dim3 grid(F_DIM / BM, F_DIM / BM, N_BATCH);   // (8, 8, 64); tj<ti blocks exit
  SecondaryStatistic_cov_wmma<<<grid, dim3(256), 0, stream>>>(x, out);
}